// ExonIntronModelPytorchCRF_40295383171088
// MI455X (gfx1250) — compile-verified
//
#include <hip/hip_runtime.h>
#include <hip/hip_bf16.h>
#include <cstddef>

// Problem constants from the reference
#define BB    16
#define TT    8192
#define DIN   4
#define HH    256
#define CC    11
#define CPAD  16          // em row padded to 16 floats for coalescing / tile stores
#define NEG_INF (-1.0e30f)

typedef __attribute__((ext_vector_type(2))) float v2f;
typedef __attribute__((ext_vector_type(8))) float v8f;

// ---------------------------------------------------------------------------
// Kernel A: fused encoder  em = relu(seq @ W_enc + b_enc) @ W_emit + b_emit
// One wave per 16-row output tile; second GEMM via V_WMMA_F32_16X16X4_F32
// (64 K-steps of 4 over H=256). The K=4 A-fragment (hidden activations) is
// recomputed in-register from the 4-wide input row (cheaper than LDS round
// trip since D_IN=4).  C/D tile layout: vgpr r -> M = r + 8*(lane/16),
// N = lane%16.  A layout: lane%16 = M, lanes 16-31 hold K=2,3.  B layout
// mirrors A with N = lane%16.
// ---------------------------------------------------------------------------
__global__ __launch_bounds__(128)
void crf_encoder_wmma_kernel(const float* __restrict__ seq,
                             const float* __restrict__ W_enc,
                             const float* __restrict__ b_enc,
                             const float* __restrict__ W_emit,
                             const float* __restrict__ b_emit,
                             float* __restrict__ em)
{
    __shared__ float s_wenc[DIN][HH];   // [d][h]
    __shared__ float s_benc[HH];
    __shared__ float s_wemit[HH][CPAD]; // [k][n], n>=11 zero-padded
    __shared__ float s_bemit[CPAD];

    const int tid = threadIdx.x;

    for (int i = tid; i < DIN * HH; i += 128)
        s_wenc[i >> 8][i & 255] = W_enc[i];
    for (int i = tid; i < HH; i += 128)
        s_benc[i] = b_enc[i];
    for (int i = tid; i < HH * CPAD; i += 128) {
        int k = i >> 4, n = i & 15;
        s_wemit[k][n] = (n < CC) ? W_emit[k * CC + n] : 0.0f;
    }
    for (int i = tid; i < CPAD; i += 128)
        s_bemit[i] = (i < CC) ? b_emit[i] : 0.0f;
    __syncthreads();

    const int wave = tid >> 5;
    const int lane = tid & 31;
    const int ml   = lane & 15;   // M for A-frag / N for B-frag / N for C-tile
    const int half = lane >> 4;   // selects K pair {0,1} vs {2,3}

    const int tile0 = (blockIdx.x * 4 + wave) * 16;   // first global row of tile

    // Per-lane input row (row M of the tile); both lane halves load the same row.
    const float4 x = ((const float4*)seq)[(size_t)tile0 + ml];

    v8f acc = {};

    #pragma unroll 4
    for (int kc = 0; kc < HH / 4; ++kc) {
        const int k0 = (kc << 2) + (half << 1);

        // A fragment: two hidden activations hidden[M][k0], hidden[M][k0+1]
        float h0 = fmaf(x.x, s_wenc[0][k0],
                   fmaf(x.y, s_wenc[1][k0],
                   fmaf(x.z, s_wenc[2][k0],
                   fmaf(x.w, s_wenc[3][k0], s_benc[k0]))));
        float h1 = fmaf(x.x, s_wenc[0][k0 + 1],
                   fmaf(x.y, s_wenc[1][k0 + 1],
                   fmaf(x.z, s_wenc[2][k0 + 1],
                   fmaf(x.w, s_wenc[3][k0 + 1], s_benc[k0 + 1]))));
        h0 = fmaxf(h0, 0.0f);
        h1 = fmaxf(h1, 0.0f);

        v2f a;  a.x = h0;               a.y = h1;
        v2f bf; bf.x = s_wemit[k0][ml]; bf.y = s_wemit[k0 + 1][ml];

        // D = A(16x4) * B(4x16) + C   -> v_wmma_f32_16x16x4_f32
        acc = __builtin_amdgcn_wmma_f32_16x16x4_f32(
                  false, a, false, bf, (short)0, acc, false, false);
    }

    // Epilogue: add emission bias, store rows of the 16x16 tile (only n<11 real)
    if (ml < CC) {
        const float be = s_bemit[ml];
        #pragma unroll
        for (int r = 0; r < 8; ++r) {
            const int m = r + (half << 3);
            em[(size_t)(tile0 + m) * CPAD + ml] = acc[r] + be;
        }
    }
}

// ---------------------------------------------------------------------------
// Kernel B: per-batch CRF.  One wave per batch element.
//   - numerator: lane-strided masked gather-sum over tags, shfl reduction
//   - forward scan: lane = destination tag c', alpha broadcast via shfl,
//     fast exp/log transcendentals; 8192 sequential steps (latency bound)
//   - denominator: wave logsumexp of alpha + end_trans
// Writes (num - den) per batch into ws_parts.
// ---------------------------------------------------------------------------
__global__ __launch_bounds__(32)
void crf_scan_kernel(const float* __restrict__ em,
                     const float* __restrict__ start_trans,
                     const float* __restrict__ trans,
                     const float* __restrict__ end_trans,
                     const int*   __restrict__ lengths,
                     const int*   __restrict__ labels,
                     float* __restrict__ ws_parts)
{
    const int b    = blockIdx.x;
    const int lane = threadIdx.x;
    const int len  = lengths[b];
    const int* lab = labels + (size_t)b * TT;
    const float* emb = em + (size_t)b * TT * CPAD;

    // ---------------- numerator (lane-strided over t) ----------------
    float part = 0.0f;
    for (int t = 1 + lane; t < TT; t += 32) {
        if (t < len) {
            int tp = lab[t - 1]; tp = (tp < 0) ? 0 : tp;
            int tc = lab[t];     tc = (tc < 0) ? 0 : tc;
            part += trans[tp * CC + tc] + emb[(size_t)t * CPAD + tc];
        }
    }
    #pragma unroll
    for (int off = 16; off > 0; off >>= 1)
        part += __shfl_down(part, off, 32);

    float num = 0.0f;
    if (lane == 0) {
        int t0 = lab[0];       t0 = (t0 < 0) ? 0 : t0;
        int tl = lab[len - 1]; tl = (tl < 0) ? 0 : tl;
        num = part + start_trans[t0] + emb[t0] + end_trans[tl];
    }

    // ---------------- forward scan ----------------
    const int  c      = lane;
    const bool active = (c < CC);

    float tr[CC];
    #pragma unroll
    for (int i = 0; i < CC; ++i)
        tr[i] = active ? trans[i * CC + c] : 0.0f;

    float alpha = active ? (start_trans[c] + emb[c]) : NEG_INF;

    for (int t = 1; t < TT; ++t) {
        const float e = active ? emb[(size_t)t * CPAD + c] : 0.0f;
        if (t + 16 < TT)
            __builtin_prefetch(&emb[(size_t)(t + 16) * CPAD + c], 0, 0);

        float v[CC];
        #pragma unroll
        for (int i = 0; i < CC; ++i)
            v[i] = __shfl(alpha, i, 32) + tr[i];

        float m = v[0];
        #pragma unroll
        for (int i = 1; i < CC; ++i) m = fmaxf(m, v[i]);

        float s = 0.0f;
        #pragma unroll
        for (int i = 0; i < CC; ++i) s += __expf(v[i] - m);

        const float nxt = m + __logf(s) + e;
        if (active && t < len) alpha = nxt;
    }

    // ---------------- denominator: logsumexp(alpha + end_trans) ----------------
    float av = active ? (alpha + end_trans[c]) : NEG_INF;
    float m = av;
    #pragma unroll
    for (int off = 16; off > 0; off >>= 1)
        m = fmaxf(m, __shfl_xor(m, off, 32));
    float sex = active ? __expf(av - m) : 0.0f;
    #pragma unroll
    for (int off = 16; off > 0; off >>= 1)
        sex += __shfl_xor(sex, off, 32);
    const float den = m + __logf(sex);

    if (lane == 0)
        ws_parts[b] = num - den;
}

// ---------------------------------------------------------------------------
// Kernel C: deterministic final reduction  out = -mean(num - den)
// ---------------------------------------------------------------------------
__global__ __launch_bounds__(32)
void crf_reduce_kernel(const float* __restrict__ ws_parts,
                       float* __restrict__ out)
{
    const int lane = threadIdx.x;
    float v = (lane < BB) ? ws_parts[lane] : 0.0f;
    #pragma unroll
    for (int off = 16; off > 0; off >>= 1)
        v += __shfl_down(v, off, 32);
    if (lane == 0)
        out[0] = -v / (float)BB;
}

// ---------------------------------------------------------------------------
extern "C" void kernel_launch(void* const* d_in, const int* in_sizes, int n_in,
                              void* d_out, int out_size, void* d_ws, size_t ws_size,
                              hipStream_t stream)
{
    (void)in_sizes; (void)n_in; (void)out_size; (void)ws_size;

    const float* seq         = (const float*)d_in[0];
    const float* W_enc       = (const float*)d_in[1];
    const float* b_enc       = (const float*)d_in[2];
    const float* W_emit      = (const float*)d_in[3];
    const float* b_emit      = (const float*)d_in[4];
    const float* start_trans = (const float*)d_in[5];
    const float* trans       = (const float*)d_in[6];
    const float* end_trans   = (const float*)d_in[7];
    const int*   lengths     = (const int*)d_in[8];
    const int*   labels      = (const int*)d_in[9];

    float* em    = (float*)d_ws;                       // (B*T, 16) padded emissions
    float* parts = em + (size_t)BB * TT * CPAD;        // 16 per-batch partials

    // A: 131072 rows, 16-row tile per wave, 4 waves per block -> 2048 blocks
    crf_encoder_wmma_kernel<<<(BB * TT) / 64, 128, 0, stream>>>(
        seq, W_enc, b_enc, W_emit, b_emit, em);

    // B: one wave per batch element
    crf_scan_kernel<<<BB, 32, 0, stream>>>(
        em, start_trans, trans, end_trans, lengths, labels, parts);

    // C: deterministic 16 -> 1 reduction
    crf_reduce_kernel<<<1, 32, 0, stream>>>(parts, (float*)d_out);
}